// GCN_2997887173232
// MI455X (gfx1250) — compile-verified
//
#include <hip/hip_runtime.h>
#include <math.h>

// ---------------------------------------------------------------------------
// GCN (3x GraphConv, norm='both', edge weights) for MI455X / gfx1250.
//  - Dense projections via V_WMMA_F32_16X16X4_F32 (full f32 precision).
//  - Edge gather/scatter via per-edge wave32 + global f32 atomics (L2-resident:
//    25.6MB node matrix << 192MB L2).
// ---------------------------------------------------------------------------

typedef __attribute__((ext_vector_type(2))) float v2f;
typedef __attribute__((ext_vector_type(8))) float v8f;

#define N_NODES 50000
#define N_EDGES 800000
#define K_DIM   128   // input feature dim for every layer

// ---------------------------------------------------------------------------
// Degree accumulation: out_deg[src]++, in_deg[dst]++
// ---------------------------------------------------------------------------
__global__ void gcn_degrees(const int* __restrict__ src, const int* __restrict__ dst,
                            float* __restrict__ outdeg, float* __restrict__ indeg) {
  int e = blockIdx.x * blockDim.x + threadIdx.x;
  if (e < N_EDGES) {
    atomicAdd(&outdeg[src[e]], 1.0f);
    atomicAdd(&indeg[dst[e]], 1.0f);
  }
}

// deg -> deg^{-1/2} with max(deg,1)
__global__ void gcn_deg_isqrt(const float* __restrict__ outdeg, const float* __restrict__ indeg,
                              float* __restrict__ odi, float* __restrict__ idi) {
  int i = blockIdx.x * blockDim.x + threadIdx.x;
  if (i < N_NODES) {
    odi[i] = rsqrtf(fmaxf(outdeg[i], 1.0f));
    idi[i] = rsqrtf(fmaxf(indeg[i], 1.0f));
  }
}

// ---------------------------------------------------------------------------
// WMMA GEMM: C[M,NOUT] = (A * odi[:,None]) @ W   with M = N_NODES (mult of 16),
// K = 128. One 16x16 tile per wave; NOUT/16 waves per block.
//   NOUT=128 -> launch 256 threads (8 waves);  NOUT=64 -> 128 threads (4 waves)
//
// Fragment layouts per CDNA5 ISA 7.12.2 (f32, wave32):
//   A 16x4:  lanes 0-15 hold M=lane, v0=K+0, v1=K+1 ; lanes 16-31: v0=K+2, v1=K+3
//   B 4x16:  lanes 0-15 hold N=lane, v0=K+0, v1=K+1 ; lanes 16-31: v0=K+2, v1=K+3
//   C 16x16: vgpr r: lanes 0-15 -> M=r, lanes 16-31 -> M=r+8, N=lane&15
// ---------------------------------------------------------------------------
template <int NOUT>
__global__ __launch_bounds__(256) void gcn_gemm_wmma(const float* __restrict__ A,
                                                     const float* __restrict__ odi,
                                                     const float* __restrict__ W,
                                                     float* __restrict__ C) {
  // +4 pad: row stride 132 words -> row r column k lands in bank (4r+k)%64,
  // so the 16 distinct rows read per half-wave hit 16 distinct banks.
  __shared__ float As[16][K_DIM + 4];

  const int tileM = blockIdx.x * 16;

  // Cooperative load of the 16x128 A stripe, pre-scaled by out_deg^{-1/2}.
  for (int idx = threadIdx.x; idx < 16 * K_DIM; idx += blockDim.x) {
    const int r = idx >> 7;       // /128
    const int c = idx & (K_DIM - 1);
    As[r][c] = A[(size_t)(tileM + r) * K_DIM + c] * odi[tileM + r];
  }
  __syncthreads();

  const int wave = threadIdx.x >> 5;
  const int lane = threadIdx.x & 31;
  const int half = lane >> 4;     // 0: K+{0,1}, 1: K+{2,3}
  const int r16  = lane & 15;
  const int n0   = wave * 16;     // launch config guarantees n0 < NOUT

  v8f acc = {};
#pragma unroll
  for (int k = 0; k < K_DIM; k += 4) {
    v2f a, b;
    a.x = As[r16][k + 2 * half];
    a.y = As[r16][k + 2 * half + 1];
    b.x = W[(size_t)(k + 2 * half)     * NOUT + n0 + r16];
    b.y = W[(size_t)(k + 2 * half + 1) * NOUT + n0 + r16];
    // (neg_a, A, neg_b, B, c_mod, C, reuse_a, reuse_b)
    acc = __builtin_amdgcn_wmma_f32_16x16x4_f32(false, a, false, b,
                                                (short)0, acc, false, false);
  }

#pragma unroll
  for (int r = 0; r < 8; ++r) {
    const int row = tileM + r + 8 * half;
    C[(size_t)row * NOUT + n0 + r16] = acc[r];
  }
}

// ---------------------------------------------------------------------------
// Edge scatter: agg[dst[e], :] += ew[e] * proj[src[e], :]
// D/4 lanes per edge, each lane moves one float4 (coalesced 512B/edge for D=128).
// ---------------------------------------------------------------------------
template <int D>
__global__ void gcn_edge_scatter(const float* __restrict__ proj,
                                 const int* __restrict__ src, const int* __restrict__ dst,
                                 const float* __restrict__ ew, float* __restrict__ agg) {
  constexpr int LPE = D / 4;  // lanes per edge
  const int gid = blockIdx.x * blockDim.x + threadIdx.x;
  const int e = gid / LPE;
  const int l = gid % LPE;
  if (e >= N_EDGES) return;

  const int s = src[e];
  const int d = dst[e];
  const float w = ew[e];

  const float4 v = ((const float4*)(proj + (size_t)s * D))[l];
  float* o = agg + (size_t)d * D + l * 4;
  atomicAdd(o + 0, v.x * w);
  atomicAdd(o + 1, v.y * w);
  atomicAdd(o + 2, v.z * w);
  atomicAdd(o + 3, v.w * w);
}

// ---------------------------------------------------------------------------
// Finalize: out = act(agg * in_deg^{-1/2} + bias)
// ---------------------------------------------------------------------------
__global__ void gcn_finalize(const float* __restrict__ agg, const float* __restrict__ idi,
                             const float* __restrict__ bias, float* __restrict__ out,
                             int D, int do_relu) {
  const int i = blockIdx.x * blockDim.x + threadIdx.x;
  if (i >= N_NODES * D) return;
  const int row = i / D;
  const int col = i % D;
  float v = agg[i] * idi[row] + bias[col];
  out[i] = do_relu ? fmaxf(v, 0.0f) : v;
}

// ---------------------------------------------------------------------------
extern "C" void kernel_launch(void* const* d_in, const int* in_sizes, int n_in,
                              void* d_out, int out_size, void* d_ws, size_t ws_size,
                              hipStream_t stream) {
  const float* feat = (const float*)d_in[0];
  const float* ew   = (const float*)d_in[1];
  const float* W1   = (const float*)d_in[2];
  const float* b1   = (const float*)d_in[3];
  const float* W2   = (const float*)d_in[4];
  const float* b2   = (const float*)d_in[5];
  const float* W3   = (const float*)d_in[6];
  const float* b3   = (const float*)d_in[7];
  const int*   src  = (const int*)d_in[8];
  const int*   dst  = (const int*)d_in[9];
  float* out = (float*)d_out;

  // Workspace layout (floats). Total = 50000*(4 + 3*128) floats ~= 77.6 MB.
  float* ws     = (float*)d_ws;
  float* outdeg = ws;                              // 50000
  float* indeg  = ws + N_NODES;                    // 50000
  float* odi    = ws + 2 * (size_t)N_NODES;        // 50000
  float* idi    = ws + 3 * (size_t)N_NODES;        // 50000
  float* h      = ws + 4 * (size_t)N_NODES;        // 50000*128
  float* proj   = h    + (size_t)N_NODES * 128;    // 50000*128
  float* agg    = proj + (size_t)N_NODES * 128;    // 50000*128

  // Degrees + normalizers.
  hipMemsetAsync(outdeg, 0, 2 * (size_t)N_NODES * sizeof(float), stream);
  gcn_degrees<<<(N_EDGES + 255) / 256, 256, 0, stream>>>(src, dst, outdeg, indeg);
  gcn_deg_isqrt<<<(N_NODES + 255) / 256, 256, 0, stream>>>(outdeg, indeg, odi, idi);

  const int gemmBlocks = N_NODES / 16;  // 50000/16 = 3125 exactly
  const int scat128Blocks = (N_EDGES * 32 + 255) / 256;  // 100000
  const int scat64Blocks  = (N_EDGES * 16 + 255) / 256;  // 50000
  const int fin128Blocks  = (N_NODES * 128 + 255) / 256; // 25000
  const int fin64Blocks   = (N_NODES * 64 + 255) / 256;  // 12500

  // ----- Layer 1: 128 -> 128, ReLU -----
  gcn_gemm_wmma<128><<<gemmBlocks, 256, 0, stream>>>(feat, odi, W1, proj);
  hipMemsetAsync(agg, 0, (size_t)N_NODES * 128 * sizeof(float), stream);
  gcn_edge_scatter<128><<<scat128Blocks, 256, 0, stream>>>(proj, src, dst, ew, agg);
  gcn_finalize<<<fin128Blocks, 256, 0, stream>>>(agg, idi, b1, h, 128, 1);

  // ----- Layer 2: 128 -> 128, ReLU -----
  gcn_gemm_wmma<128><<<gemmBlocks, 256, 0, stream>>>(h, odi, W2, proj);
  hipMemsetAsync(agg, 0, (size_t)N_NODES * 128 * sizeof(float), stream);
  gcn_edge_scatter<128><<<scat128Blocks, 256, 0, stream>>>(proj, src, dst, ew, agg);
  gcn_finalize<<<fin128Blocks, 256, 0, stream>>>(agg, idi, b2, h, 128, 1);

  // ----- Layer 3: 128 -> 64, no activation, write d_out -----
  gcn_gemm_wmma<64><<<gemmBlocks, 128, 0, stream>>>(h, odi, W3, proj);
  hipMemsetAsync(agg, 0, (size_t)N_NODES * 64 * sizeof(float), stream);
  gcn_edge_scatter<64><<<scat64Blocks, 256, 0, stream>>>(proj, src, dst, ew, agg);
  gcn_finalize<<<fin64Blocks, 256, 0, stream>>>(agg, idi, b3, out, 64, 0);
}